// PatchedPointsRenderer_47047071761133
// MI455X (gfx1250) — compile-verified
//
#include <hip/hip_runtime.h>

// PatchedPointsRenderer for MI455X (gfx1250, wave32).
//
// Inputs (setup_inputs order):
//   d_in[0] idx      (B,H,W,K) int32, -1 = background
//   d_in[1] dists    (B,H,W,K) float32 (squared distances)
//   d_in[2] radii    (P,)      float32
//   d_in[3] features (C,P)     float32
// Output: images (B,H,W,C) float32, C == 16, K == 8.
//
// Plan:
//   1) transpose features (C,P) -> featT (P,C) in d_ws so each point's
//      feature vector is one contiguous 64B line (L2-resident, 12.8 MB).
//   2) one wave32 per pixel: compute the 8 alphas (exclusive cumprod via
//      lane shuffles), then do the KxC contraction with two chained
//      V_WMMA_F32_16X16X4_F32 ops. B operand (4x16, lane = channel) is
//      filled by perfectly coalesced 64B gathers from featT; A operand
//      broadcasts alpha down all 16 rows, so D row 0 is the pixel's
//      16-channel output.

typedef float v2f __attribute__((ext_vector_type(2)));
typedef float v8f __attribute__((ext_vector_type(8)));

#define KF 8
#define CF 16

// ---------------------------------------------------------------------------
// Kernel 1: (C,P) -> (P,C) transpose, LDS-tiled.
// Reads coalesced along P, stages through LDS (pad 257 -> conflict-free on
// 64 banks since 257 % 64 == 1), writes coalesced along the (P,C) output.
// ---------------------------------------------------------------------------
__global__ void feat_transpose_kernel(const float* __restrict__ feat,
                                      float* __restrict__ featT, int P) {
  __shared__ float tile[CF][257];
  const int t = threadIdx.x;           // 0..255
  const int p0 = blockIdx.x * 256;
  const int p = p0 + t;
#pragma unroll
  for (int c = 0; c < CF; ++c)
    tile[c][t] = (p < P) ? feat[(long)c * P + p] : 0.0f;
  __syncthreads();
#pragma unroll
  for (int i = 0; i < CF; ++i) {
    const int j = i * 256 + t;         // 0..4095, consecutive across lanes
    const int pp = j >> 4;             // local point
    const int c = j & 15;              // channel
    if (p0 + pp < P) featT[(long)(p0 + pp) * CF + c] = tile[c][pp];
  }
}

// ---------------------------------------------------------------------------
// Kernel 2: one wave per pixel, alpha compositing + WMMA contraction.
// ---------------------------------------------------------------------------
__global__ void composite_wmma_kernel(const int* __restrict__ idx,
                                      const float* __restrict__ dists,
                                      const float* __restrict__ radii,
                                      const float* __restrict__ featT,
                                      float* __restrict__ out, int npix) {
  const int lane = threadIdx.x & 31;
  const int wave =
      blockIdx.x * ((int)blockDim.x >> 5) + ((int)threadIdx.x >> 5);
  if (wave >= npix) return;  // wave-uniform; EXEC stays all-ones for WMMA

  const long base = (long)wave * KF;

  // Lanes 0..7 own slots 0..7 (lanes 8..31 hold redundant copies of slot
  // lane&7; all 8 distinct loads land in one 32B span -> one cache line).
  const int s = lane & 7;
  const int id = idx[base + s];
  const float dd = dists[base + s];
  const bool valid = id >= 0;
  const int sid = valid ? id : 0;
  const float r = radii[sid];
  const float w = valid ? (1.0f - dd / (r * r)) : 0.0f;

  // Exclusive cumprod of (1-w): replicate the 8-slot scan in every lane via
  // wave32 shuffles (tiny serial chain, identical across lanes).
  float alpha[KF];
  int sids[KF];
  float t = 1.0f;
#pragma unroll
  for (int j = 0; j < KF; ++j) {
    const float wj = __shfl(w, j, 32);
    sids[j] = __shfl(sid, j, 32);
    alpha[j] = wj * t;
    t *= (1.0f - wj);
  }

  // WMMA operand build. ISA 16x16x4 F32 layout:
  //   A (16x4): VGPR0 = K0 (lanes 0-15) / K2 (lanes 16-31); VGPR1 = K1 / K3
  //   B (4x16): same K striping; lane%16 = N = channel
  //   D (16x16): VGPR0 = M=0 (lanes 0-15, N=lane) / M=8 (lanes 16-31)
  const int c = lane & 15;   // channel for B/D
  const int hi = lane >> 4;  // lane half selects even/odd K pair

  v8f acc = {};  // C = 0

  {  // slots 0..3
    const int k0 = hi ? 2 : 0;
    const int k1 = hi ? 3 : 1;
    v2f A, Bv;
    A.x = alpha[k0];
    A.y = alpha[k1];
    Bv.x = featT[(long)sids[k0] * CF + c];  // coalesced 64B per half-wave
    Bv.y = featT[(long)sids[k1] * CF + c];
    acc = __builtin_amdgcn_wmma_f32_16x16x4_f32(false, A, false, Bv, (short)0,
                                                acc, false, false);
  }
  {  // slots 4..7
    const int k0 = hi ? 6 : 4;
    const int k1 = hi ? 7 : 5;
    v2f A, Bv;
    A.x = alpha[k0];
    A.y = alpha[k1];
    Bv.x = featT[(long)sids[k0] * CF + c];
    Bv.y = featT[(long)sids[k1] * CF + c];
    acc = __builtin_amdgcn_wmma_f32_16x16x4_f32(false, A, false, Bv, (short)0,
                                                acc, false, false);
  }

  // Every D row equals the pixel's output vector (A rows identical), so
  // row M=0 (acc[0] on lanes 0-15) is the complete 16-channel result.
  if (lane < 16) out[(long)wave * CF + lane] = acc[0];
}

// ---------------------------------------------------------------------------
extern "C" void kernel_launch(void* const* d_in, const int* in_sizes, int n_in,
                              void* d_out, int out_size, void* d_ws,
                              size_t ws_size, hipStream_t stream) {
  const int* idx = (const int*)d_in[0];
  const float* dists = (const float*)d_in[1];
  const float* radii = (const float*)d_in[2];
  const float* features = (const float*)d_in[3];
  float* out = (float*)d_out;

  const int P = in_sizes[2];          // 200000
  const int npix = out_size / CF;     // B*H*W = 1,048,576

  float* featT = (float*)d_ws;        // P*16 floats = 12.8 MB scratch

  const int tblocks = (P + 255) / 256;
  feat_transpose_kernel<<<tblocks, 256, 0, stream>>>(features, featT, P);

  const int waves_per_block = 256 / 32;  // 8 waves (wave32)
  const int cblocks = (npix + waves_per_block - 1) / waves_per_block;
  composite_wmma_kernel<<<cblocks, 256, 0, stream>>>(idx, dists, radii, featT,
                                                     out, npix);
}